// TitansConv_3393024164107
// MI455X (gfx1250) — compile-verified
//
#include <hip/hip_runtime.h>

// Depthwise causal conv1d (K=4) + bias + SiLU, layout [B,S,H,D] -> [B,H,S,D].
// Memory-bound (AI ~1.25 F/B): optimized for 23.3 TB/s HBM streaming on MI455X.
//  - 128-bit coalesced loads/stores, one wave == one head for both in and out
//  - non-temporal cache policy (stream-once footprint 268MB > 192MB L2)
//  - sliding 4-deep window in registers: every input byte read exactly once
//  - gfx1250 async global->LDS prefetch FIFO (ASYNCcnt-tracked)

typedef float v4f __attribute__((ext_vector_type(4)));
typedef int   v4i __attribute__((ext_vector_type(4)));

namespace {
constexpr int BB = 4;
constexpr int SS = 4096;
constexpr int HH = 16;
constexpr int DD = 128;
constexpr int CC = HH * DD;     // 2048 channels
constexpr int KK = 4;
constexpr int TPB = 256;        // 8 waves (wave32)
constexpr int CPB = TPB * 4;    // 1024 channels per block (4 per thread, b128)
constexpr int CG  = CC / CPB;   // 2 channel groups
constexpr int SC  = 64;         // s-rows per block
constexpr int NCHUNK = SS / SC; // 64  -> grid = 2*4*64 = 512 blocks
constexpr int NSTAGE = 8;       // async FIFO depth (32 KB LDS)
}

__device__ __forceinline__ float silu_f(float v) {
  return v * __builtin_amdgcn_rcpf(1.0f + __expf(-v));
}
__device__ __forceinline__ v4f nt_load4(const float* p) {
  return __builtin_nontemporal_load((const v4f*)p);
}
__device__ __forceinline__ void nt_store4(float* p, v4f v) {
  __builtin_nontemporal_store(v, (v4f*)p);
}

// Shared per-thread setup: channel mapping, transposed taps, pointers.
struct Ctx {
  const float* xin;   // + s*CC steps
  float* yout;        // + s*DD steps
  v4f w0, w1, w2, w3, bz;
  int s0;
};

__device__ __forceinline__ Ctx make_ctx(const float* __restrict__ x,
                                        const float* __restrict__ w,
                                        const float* __restrict__ bias,
                                        float* __restrict__ out) {
  Ctx c;
  const int gid   = blockIdx.x;
  const int cg    = gid % CG;
  const int b     = (gid / CG) % BB;
  const int chunk = gid / (CG * BB);
  const int t  = threadIdx.x;
  const int c0 = cg * CPB + t * 4;
  const int h  = c0 / DD;           // 32 lanes * 4ch == 128 == D: one head/wave
  const int d0 = c0 % DD;

  // conv_weight is [C][4] row-major: load 4 rows, transpose to per-tap vectors
  v4f r0 = *(const v4f*)(w + (size_t)(c0 + 0) * KK);
  v4f r1 = *(const v4f*)(w + (size_t)(c0 + 1) * KK);
  v4f r2 = *(const v4f*)(w + (size_t)(c0 + 2) * KK);
  v4f r3 = *(const v4f*)(w + (size_t)(c0 + 3) * KK);
  c.w0 = (v4f){r0.x, r1.x, r2.x, r3.x};
  c.w1 = (v4f){r0.y, r1.y, r2.y, r3.y};
  c.w2 = (v4f){r0.z, r1.z, r2.z, r3.z};
  c.w3 = (v4f){r0.w, r1.w, r2.w, r3.w};
  c.bz = *(const v4f*)(bias + c0);

  c.xin  = x + (size_t)b * SS * CC + c0;
  c.yout = out + ((size_t)(b * HH + h) * SS) * DD + d0;
  c.s0   = chunk * SC;
  return c;
}

__device__ __forceinline__ void load_halo(const Ctx& c, v4f& xm3, v4f& xm2, v4f& xm1) {
  xm3 = (v4f){0.f, 0.f, 0.f, 0.f};
  xm2 = xm3; xm1 = xm3;
  if (c.s0 != 0) {  // SC >= 3, so all three rows exist when s0 > 0
    xm3 = *(const v4f*)(c.xin + (size_t)(c.s0 - 3) * CC);
    xm2 = *(const v4f*)(c.xin + (size_t)(c.s0 - 2) * CC);
    xm1 = *(const v4f*)(c.xin + (size_t)(c.s0 - 1) * CC);
  }
}

__device__ __forceinline__ void step(const Ctx& c, int s, v4f xc,
                                     v4f& xm3, v4f& xm2, v4f& xm1) {
  v4f y = c.bz + c.w0 * xm3 + c.w1 * xm2 + c.w2 * xm1 + c.w3 * xc;
  y.x = silu_f(y.x); y.y = silu_f(y.y); y.z = silu_f(y.z); y.w = silu_f(y.w);
  nt_store4(c.yout + (size_t)s * DD, y);
  xm3 = xm2; xm2 = xm1; xm1 = xc;
}

// ---------------- Direct register-streaming kernel ----------------
__global__ __launch_bounds__(TPB) void conv_silu_direct(
    const float* __restrict__ x, const float* __restrict__ w,
    const float* __restrict__ bias, float* __restrict__ out) {
  Ctx c = make_ctx(x, w, bias, out);
  v4f xm3, xm2, xm1;
  load_halo(c, xm3, xm2, xm1);
  for (int i = 0; i < SC; i += 8) {
    int ps = c.s0 + i + 24; if (ps > SS - 1) ps = SS - 1;
    __builtin_prefetch(c.xin + (size_t)ps * CC, 0, 1);  // global_prefetch_b8
#pragma unroll
    for (int u = 0; u < 8; ++u) {
      const int s = c.s0 + i + u;
      v4f xc = nt_load4(c.xin + (size_t)s * CC);
      step(c, s, xc, xm3, xm2, xm1);
    }
  }
}

// ---------------- Async global->LDS FIFO kernel (gfx1250) ----------------
#if __has_builtin(__builtin_amdgcn_global_load_async_to_lds_b128)
#define HAVE_ASYNC_LDS 1

typedef __attribute__((address_space(1))) v4i gv4i;  // global int4
typedef __attribute__((address_space(3))) v4i lv4i;  // LDS int4

__device__ __forceinline__ void async_copy16(const float* g, float* l) {
  // Signature per clang diagnostic: (int4 AS1*, int4 AS3*, Imm offset, Imm cpol)
  __builtin_amdgcn_global_load_async_to_lds_b128((gv4i*)g, (lv4i*)l, 0, 0);
}

template <int N>
__device__ __forceinline__ void wait_async() {
#if __has_builtin(__builtin_amdgcn_s_wait_asynccnt)
  __builtin_amdgcn_s_wait_asynccnt(N);   // N is an ICE via template param
#else
  asm volatile("s_wait_asynccnt %0" ::"i"(N) : "memory");
#endif
}
__device__ __forceinline__ void wait_ds0() {
#if __has_builtin(__builtin_amdgcn_s_wait_dscnt)
  __builtin_amdgcn_s_wait_dscnt(0);
#else
  asm volatile("s_wait_dscnt 0x0" ::: "memory");
#endif
}

__global__ __launch_bounds__(TPB) void conv_silu_async(
    const float* __restrict__ x, const float* __restrict__ w,
    const float* __restrict__ bias, float* __restrict__ out) {
  __shared__ float lds[NSTAGE * CPB];           // 32 KB prefetch FIFO
  Ctx c = make_ctx(x, w, bias, out);
  float* slot = &lds[threadIdx.x * 4];          // stage st at slot + st*CPB
  v4f xm3, xm2, xm1;
  load_halo(c, xm3, xm2, xm1);

  // Prime the FIFO: each lane DMAs its own 16B; no cross-thread sharing,
  // so per-wave ASYNCcnt ordering suffices (no barriers).
#pragma unroll
  for (int st = 0; st < NSTAGE; ++st)
    async_copy16(c.xin + (size_t)(c.s0 + st) * CC, slot + st * CPB);

  for (int i = 0; i < SC - NSTAGE; ++i) {
    wait_async<NSTAGE - 1>();                   // oldest row landed (in-order)
    const int st = i & (NSTAGE - 1);
    v4f xc = *(const v4f*)(slot + st * CPB);    // ds_load_b128
    wait_ds0();                                 // slot free before refill
    async_copy16(c.xin + (size_t)(c.s0 + i + NSTAGE) * CC, slot + st * CPB);
    step(c, c.s0 + i, xc, xm3, xm2, xm1);
  }
  wait_async<0>();                              // drain tail
#pragma unroll
  for (int i = SC - NSTAGE; i < SC; ++i) {
    const int st = i & (NSTAGE - 1);
    v4f xc = *(const v4f*)(slot + st * CPB);
    step(c, c.s0 + i, xc, xm3, xm2, xm1);
  }
}
#endif  // async builtin

extern "C" void kernel_launch(void* const* d_in, const int* in_sizes, int n_in,
                              void* d_out, int out_size, void* d_ws, size_t ws_size,
                              hipStream_t stream) {
  const float* x    = (const float*)d_in[0];  // [B,S,H,D]
  const float* w    = (const float*)d_in[1];  // [C,4]
  const float* bias = (const float*)d_in[2];  // [C]
  float* out = (float*)d_out;                 // [B,H,S,D]
  dim3 grid(CG * BB * NCHUNK), block(TPB);
#ifdef HAVE_ASYNC_LDS
  conv_silu_async<<<grid, block, 0, stream>>>(x, w, bias, out);
#else
  conv_silu_direct<<<grid, block, 0, stream>>>(x, w, bias, out);
#endif
  (void)in_sizes; (void)n_in; (void)out_size; (void)d_ws; (void)ws_size;
}